// DetModel_38603166056674
// MI455X (gfx1250) — compile-verified
//
#include <hip/hip_runtime.h>

typedef __attribute__((ext_vector_type(2))) float v2f;
typedef __attribute__((ext_vector_type(8))) float v8f;

#define TWO_RX 64
#define TWO_TX 64
#define HID    256

// LDS strides padded so the two half-wave fragment gathers hit disjoint bank sets.
#define SH_STRIDE   80   // H tile (64x64): banks (r*80+c)%64 -> halves offset by 32
#define SHTH_STRIDE 65   // HtH: column broadcast in GD loop is conflict-free
#define SZ_STRIDE   68   // Z tile (16x64): banks (m*68+k)%64 = (4m+k)%64
#define SHH_STRIDE  260  // h / h_new tiles (16x256): same 4m+k property

// ---------------------------------------------------------------------------
// Kernel 1: per-batch  HtH = H^T H (WMMA f32), Hty, 20 GD iterations -> z
// ---------------------------------------------------------------------------
__global__ __launch_bounds__(256) void gd_kernel(
    const float* __restrict__ y, const float* __restrict__ H,
    const float* __restrict__ x0, const float* __restrict__ step,
    const int* __restrict__ iters, float* __restrict__ z_out)
{
    __shared__ float sH[64 * SH_STRIDE];
    __shared__ float sHtH[64 * SHTH_STRIDE];
    __shared__ float sy[64];
    __shared__ float sHty[64];
    __shared__ float sx[64];

    const int b    = blockIdx.x;
    const int tid  = threadIdx.x;
    const int lane = tid & 31;
    const int wave = tid >> 5;
    const int m    = lane & 15;   // row (A) / col (B) index inside 16-wide tile
    const int kh   = lane >> 4;   // half-wave selects K pair

    // ---- stage H (64x64 f32, 16 KB) into LDS, float4-vectorized ----
    {
        const float4* Hg = (const float4*)(H + (size_t)b * TWO_RX * TWO_TX);
        for (int i = tid; i < 64 * 16; i += 256) {   // 1024 float4s
            int row = i >> 4;
            int c4  = (i & 15) << 2;
            float4 v = Hg[i];
            float* p = &sH[row * SH_STRIDE + c4];
            p[0] = v.x; p[1] = v.y; p[2] = v.z; p[3] = v.w;
        }
    }
    if (tid < 64) {
        sy[tid] = y[(size_t)b * TWO_RX + tid];
        sx[tid] = x0[(size_t)b * TWO_TX + tid];
    }
    __syncthreads();

    // ---- HtH[s][t] = sum_r H[r][s]*H[r][t] : 4x4 grid of 16x16 WMMA tiles ----
    for (int t = wave; t < 16; t += 8) {           // 2 tiles per wave, wave-uniform
        const int ti = t >> 2, tj = t & 3;
        v8f c = {};
        #pragma unroll
        for (int kk = 0; kk < 16; ++kk) {          // K = 64 in steps of 4
            const int r0 = kk * 4 + kh * 2;
            v2f a, bb;
            a.x  = sH[(r0    ) * SH_STRIDE + ti * 16 + m];
            a.y  = sH[(r0 + 1) * SH_STRIDE + ti * 16 + m];
            bb.x = sH[(r0    ) * SH_STRIDE + tj * 16 + m];
            bb.y = sH[(r0 + 1) * SH_STRIDE + tj * 16 + m];
            c = __builtin_amdgcn_wmma_f32_16x16x4_f32(false, a, false, bb,
                                                      (short)0, c, false, false);
        }
        #pragma unroll
        for (int r = 0; r < 8; ++r) {
            const int row = ti * 16 + r + kh * 8;
            const int col = tj * 16 + m;
            sHtH[row * SHTH_STRIDE + col] = c[r];
        }
    }
    __syncthreads();

    // ---- Hty[t] = sum_r H[r][t]*y[r]  (VALU; conflict-free column reads) ----
    if (tid < 64) {
        float acc = 0.f;
        #pragma unroll 8
        for (int r = 0; r < 64; ++r)
            acc = __builtin_fmaf(sH[r * SH_STRIDE + tid], sy[r], acc);
        sHty[tid] = acc;
    }
    __syncthreads();

    // ---- GD: x += 2*step*(Hty - HtH x), serial over iterations ----
    const float ts2  = 2.0f * step[0];
    const int   n_it = iters[0];
    for (int it = 0; it < n_it; ++it) {
        float xn = 0.f;
        if (tid < 64) {
            float acc = 0.f;
            #pragma unroll 8
            for (int s = 0; s < 64; ++s)
                acc = __builtin_fmaf(sHtH[tid * SHTH_STRIDE + s], sx[s], acc);
            xn = sx[tid] + ts2 * (sHty[tid] - acc);
        }
        __syncthreads();
        if (tid < 64) sx[tid] = xn;
        __syncthreads();
    }

    if (tid < 64) z_out[(size_t)b * TWO_TX + tid] = sx[tid];
}

// ---------------------------------------------------------------------------
// Kernel 2: batched RNN cell + projection as WMMA GEMMs over 16-row tiles
//   h_new = relu(Z W_ih^T + h W_hh^T + b_ih + b_hh);  x_out = h_new w_x^T + b_x
// ---------------------------------------------------------------------------
__global__ __launch_bounds__(256) void rnn_kernel(
    const float* __restrict__ z, const float* __restrict__ h_in,
    const float* __restrict__ W_ih, const float* __restrict__ W_hh,
    const float* __restrict__ b_ih, const float* __restrict__ b_hh,
    const float* __restrict__ w_x, const float* __restrict__ b_x,
    float* __restrict__ out_x, float* __restrict__ out_h)
{
    __shared__ float sZ[16 * SZ_STRIDE];
    __shared__ float sh[16 * SHH_STRIDE];
    __shared__ float sHn[16 * SHH_STRIDE];

    const size_t rowbase = (size_t)blockIdx.x * 16;
    const int tid  = threadIdx.x;
    const int lane = tid & 31;
    const int wave = tid >> 5;
    const int m    = lane & 15;
    const int kh   = lane >> 4;

    // ---- stage Z tile (16x64) and h tile (16x256) ----
    {
        const float4* zg = (const float4*)(z + rowbase * TWO_TX);
        int row = tid >> 4, c4 = (tid & 15) << 2;      // 256 float4s exactly
        float4 v = zg[tid];
        float* p = &sZ[row * SZ_STRIDE + c4];
        p[0] = v.x; p[1] = v.y; p[2] = v.z; p[3] = v.w;
    }
    {
        const float4* hg = (const float4*)(h_in + rowbase * HID);
        for (int i = tid; i < 16 * 64; i += 256) {     // 1024 float4s
            int row = i >> 6, c4 = (i & 63) << 2;
            float4 v = hg[i];
            float* p = &sh[row * SHH_STRIDE + c4];
            p[0] = v.x; p[1] = v.y; p[2] = v.z; p[3] = v.w;
        }
    }
    __syncthreads();

    // ---- h_new: 16 column tiles of 16; wave handles tiles {wave, wave+8} ----
    for (int nt = wave; nt < 16; nt += 8) {
        const int cb = nt * 16;
        v8f c = {};
        #pragma unroll
        for (int kk = 0; kk < 16; ++kk) {              // Z @ W_ih^T, K = 64
            const int k0 = kk * 4 + kh * 2;
            v2f a, bb;
            a.x = sZ[m * SZ_STRIDE + k0];
            a.y = sZ[m * SZ_STRIDE + k0 + 1];
            const float* wp = W_ih + (size_t)(cb + m) * TWO_TX + k0; // B[k][n]=W_ih[n][k]
            bb.x = wp[0]; bb.y = wp[1];
            c = __builtin_amdgcn_wmma_f32_16x16x4_f32(false, a, false, bb,
                                                      (short)0, c, false, false);
        }
        #pragma unroll 8
        for (int kk = 0; kk < 64; ++kk) {              // h @ W_hh^T, K = 256
            const int k0 = kk * 4 + kh * 2;
            v2f a, bb;
            a.x = sh[m * SHH_STRIDE + k0];
            a.y = sh[m * SHH_STRIDE + k0 + 1];
            const float* wp = W_hh + (size_t)(cb + m) * HID + k0;    // L2-resident
            bb.x = wp[0]; bb.y = wp[1];
            c = __builtin_amdgcn_wmma_f32_16x16x4_f32(false, a, false, bb,
                                                      (short)0, c, false, false);
        }
        const float bias = b_ih[cb + m] + b_hh[cb + m];
        #pragma unroll
        for (int r = 0; r < 8; ++r) {
            const int row = r + kh * 8;                // batch row inside tile
            float v = c[r] + bias;
            v = v > 0.f ? v : 0.f;                     // relu
            sHn[row * SHH_STRIDE + cb + m] = v;
            out_h[(rowbase + row) * HID + cb + m] = v;
        }
    }
    __syncthreads();

    // ---- x_out = h_new @ w_x^T + b_x : 4 column tiles, waves 0..3 ----
    if (wave < 4) {                                    // wave-uniform branch
        const int cb = wave * 16;
        v8f c = {};
        #pragma unroll 8
        for (int kk = 0; kk < 64; ++kk) {              // K = 256
            const int k0 = kk * 4 + kh * 2;
            v2f a, bb;
            a.x = sHn[m * SHH_STRIDE + k0];
            a.y = sHn[m * SHH_STRIDE + k0 + 1];
            const float* wp = w_x + (size_t)(cb + m) * HID + k0;     // B[k][n]=w_x[n][k]
            bb.x = wp[0]; bb.y = wp[1];
            c = __builtin_amdgcn_wmma_f32_16x16x4_f32(false, a, false, bb,
                                                      (short)0, c, false, false);
        }
        const float bias = b_x[cb + m];
        #pragma unroll
        for (int r = 0; r < 8; ++r) {
            const int row = r + kh * 8;
            out_x[(rowbase + row) * TWO_TX + cb + m] = c[r] + bias;
        }
    }
}

// ---------------------------------------------------------------------------
extern "C" void kernel_launch(void* const* d_in, const int* in_sizes, int n_in,
                              void* d_out, int out_size, void* d_ws, size_t ws_size,
                              hipStream_t stream)
{
    const float* y    = (const float*)d_in[0];
    const float* H    = (const float*)d_in[1];
    const float* x0   = (const float*)d_in[2];
    const float* h    = (const float*)d_in[3];
    const float* stp  = (const float*)d_in[4];
    const float* W_ih = (const float*)d_in[5];
    const float* W_hh = (const float*)d_in[6];
    const float* b_ih = (const float*)d_in[7];
    const float* b_hh = (const float*)d_in[8];
    const float* w_x  = (const float*)d_in[9];
    const float* b_x  = (const float*)d_in[10];
    const int*   itr  = (const int*)d_in[11];

    const int B = in_sizes[0] / TWO_RX;                // 16384

    float* z_ws  = (float*)d_ws;                       // B*64 f32 = 4 MB scratch
    float* out_x = (float*)d_out;                      // (B,64) then (B,256)
    float* out_h = out_x + (size_t)B * TWO_TX;

    gd_kernel<<<B, 256, 0, stream>>>(y, H, x0, stp, itr, z_ws);
    rnn_kernel<<<B / 16, 256, 0, stream>>>(z_ws, h, W_ih, W_hh, b_ih, b_hh,
                                           w_x, b_x, out_x, out_h);
}